// AttnDecoderRNN_23149873725851
// MI455X (gfx1250) — compile-verified
//
#include <hip/hip_runtime.h>
#include <hip/hip_bf16.h>
#include <math.h>

typedef __attribute__((ext_vector_type(2))) float v2f;
typedef __attribute__((ext_vector_type(8))) float v8f;

#define HIDDEN 1024
#define VOCAB  50257
#define MAXLEN 12

// ---------------------------------------------------------------------------
// Kernel 1: embedding lookup + attention logits + softmax + context vector.
// Writes xc[0..1023]=embedded, xc[1024..2047]=attn_applied, attn weights to out.
// ---------------------------------------------------------------------------
__global__ void attn_kernel(const int* __restrict__ tok,
                            const float* __restrict__ h0,
                            const float* __restrict__ enc,
                            const float* __restrict__ emb,
                            const float* __restrict__ attn_W,
                            const float* __restrict__ attn_b,
                            float* __restrict__ xc,
                            float* __restrict__ attnw_out) {
  __shared__ float red[256];
  __shared__ float wsm[MAXLEN];
  const int t = threadIdx.x;
  const int token = tok[0];  // low dword of the int64 index (little-endian)
  const float* erow = emb + (size_t)token * HIDDEN;

  for (int i = t; i < HIDDEN; i += 256) xc[i] = erow[i];

  // attention logits: dot([emb, h0], attn_W[j,:]) + b[j]
  for (int j = 0; j < MAXLEN; ++j) {
    const float* wr = attn_W + j * (2 * HIDDEN);
    float p = 0.f;
    for (int k = t; k < HIDDEN; k += 256)
      p += erow[k] * wr[k] + h0[k] * wr[HIDDEN + k];
    red[t] = p; __syncthreads();
    for (int s = 128; s > 0; s >>= 1) {
      if (t < s) red[t] += red[t + s];
      __syncthreads();
    }
    if (t == 0) wsm[j] = red[0] + attn_b[j];
    __syncthreads();
  }

  if (t == 0) {
    float mx = wsm[0];
    for (int j = 1; j < MAXLEN; ++j) mx = fmaxf(mx, wsm[j]);
    float s = 0.f;
    for (int j = 0; j < MAXLEN; ++j) { wsm[j] = expf(wsm[j] - mx); s += wsm[j]; }
    float inv = 1.f / s;
    for (int j = 0; j < MAXLEN; ++j) { wsm[j] *= inv; attnw_out[j] = wsm[j]; }
  }
  __syncthreads();

  // context = attn_weights @ encoder_outputs
  for (int h = t; h < HIDDEN; h += 256) {
    float a = 0.f;
    #pragma unroll
    for (int j = 0; j < MAXLEN; ++j) a += wsm[j] * enc[j * HIDDEN + h];
    xc[HIDDEN + h] = a;
  }
}

// ---------------------------------------------------------------------------
// WMMA fp32 GEMV: out[m] = act( dot(W[m,:K], x) + b0[m] + b1[m] (+ out[m]) )
// One wave per 16-row tile. A = 16x4 tile of W, B = x[k..k+3] broadcast into
// all 16 columns -> every column of the 16x16 accumulator holds the 16 dots.
// x is staged in LDS (K <= 2048).
// ---------------------------------------------------------------------------
__global__ void wmma_gemv_kernel(const float* __restrict__ W,
                                 const float* __restrict__ x,
                                 const float* __restrict__ b0,
                                 const float* __restrict__ b1,
                                 float* __restrict__ out,
                                 int M, int K, int accumulate, int relu) {
  __shared__ float xs[2048];
  for (int i = threadIdx.x; i < K; i += blockDim.x) xs[i] = x[i];
  __syncthreads();

  const int lane = threadIdx.x & 31;
  const int waveInBlock = threadIdx.x >> 5;
  const int wavesPerBlock = blockDim.x >> 5;
  const int tile = blockIdx.x * wavesPerBlock + waveInBlock;
  const bool active = (tile * 16) < M;
  const int m0 = (active ? tile : 0) * 16;

  // A-matrix layout (32-bit, 16x4): lanes 0-15 -> M=lane, K={0,1};
  //                                 lanes 16-31 -> M=lane-16, K={2,3}.
  const int mrow = m0 + (lane & 15);
  const int row = (mrow < M) ? mrow : (M - 1);     // clamp: safe loads only
  const int khalf = (lane >> 4) << 1;              // 0 or 2
  const float* wrow = W + (size_t)row * K + khalf;

  v8f acc = {};
  #pragma unroll 4
  for (int k = 0; k < K; k += 4) {
    v2f a = *(const v2f*)(wrow + k);               // global_load_b64, 8B aligned
    v2f b;
    b.x = xs[k + khalf];
    b.y = xs[k + khalf + 1];
    acc = __builtin_amdgcn_wmma_f32_16x16x4_f32(
        /*neg_a=*/false, a, /*neg_b=*/false, b,
        /*c_mod=*/(short)0, acc, /*reuse_a=*/false, /*reuse_b=*/false);
  }

  // D layout: lane column = lane&15; VGPR r holds row r + 8*(lane>>4).
  // All columns identical -> lanes 0 and 16 emit the 16 results.
  if (active && (lane & 15) == 0) {
    const int base = m0 + ((lane >> 4) << 3);
    #pragma unroll
    for (int r = 0; r < 8; ++r) {
      const int m = base + r;
      if (m < M) {
        float v = acc[r];
        if (b0) v += b0[m];
        if (b1) v += b1[m];
        if (accumulate) v += out[m];
        if (relu) v = fmaxf(v, 0.f);
        out[m] = v;
      }
    }
  }
}

// ---------------------------------------------------------------------------
// LSTM cell elementwise (gate order i, f, g, o)
// ---------------------------------------------------------------------------
__global__ void lstm_kernel(const float* __restrict__ gates,
                            const float* __restrict__ c0,
                            float* __restrict__ h_out,
                            float* __restrict__ c_out,
                            float* __restrict__ h_ws) {
  const int i = blockIdx.x * blockDim.x + threadIdx.x;
  if (i >= HIDDEN) return;
  const float ig = gates[i];
  const float fg = gates[HIDDEN + i];
  const float gg = gates[2 * HIDDEN + i];
  const float og = gates[3 * HIDDEN + i];
  const float si = 1.f / (1.f + expf(-ig));
  const float sf = 1.f / (1.f + expf(-fg));
  const float so = 1.f / (1.f + expf(-og));
  const float c = sf * c0[i] + si * tanhf(gg);
  const float h = so * tanhf(c);
  c_out[i] = c;
  h_out[i] = h;
  h_ws[i] = h;
}

// ---------------------------------------------------------------------------
// Streaming log-sum-exp: per-block (max, scaled-sum) partials in one pass.
// ---------------------------------------------------------------------------
__global__ void lse_partial_kernel(const float* __restrict__ logits, int n,
                                   float* __restrict__ pm, float* __restrict__ ps) {
  __shared__ float sm[256], ss[256];
  const int t = threadIdx.x;
  const int stride = gridDim.x * blockDim.x;
  float m = -INFINITY, s = 0.f;
  for (int i = blockIdx.x * blockDim.x + t; i < n; i += stride) {
    const float v = logits[i];
    if (v > m) { s = s * expf(m - v) + 1.f; m = v; }
    else       { s += expf(v - m); }
  }
  sm[t] = m; ss[t] = s; __syncthreads();
  for (int k = 128; k > 0; k >>= 1) {
    if (t < k) {
      const float ma = sm[t], mb = sm[t + k];
      const float mm = fmaxf(ma, mb);
      float sn = 0.f;
      if (ma > -INFINITY) sn += ss[t] * expf(ma - mm);
      if (mb > -INFINITY) sn += ss[t + k] * expf(mb - mm);
      sm[t] = mm; ss[t] = sn;
    }
    __syncthreads();
  }
  if (t == 0) { pm[blockIdx.x] = sm[0]; ps[blockIdx.x] = ss[0]; }
}

__global__ void lse_final_kernel(const float* __restrict__ pm,
                                 const float* __restrict__ ps, int n,
                                 float* __restrict__ lse) {
  if (threadIdx.x == 0) {
    float m = -INFINITY;
    for (int i = 0; i < n; ++i) m = fmaxf(m, pm[i]);
    float s = 0.f;
    for (int i = 0; i < n; ++i) s += ps[i] * expf(pm[i] - m);
    *lse = m + logf(s);
  }
}

__global__ void sub_lse_kernel(float* __restrict__ logits,
                               const float* __restrict__ lse, int n) {
  const int i = blockIdx.x * blockDim.x + threadIdx.x;
  if (i < n) logits[i] -= *lse;
}

// ---------------------------------------------------------------------------
extern "C" void kernel_launch(void* const* d_in, const int* in_sizes, int n_in,
                              void* d_out, int out_size, void* d_ws, size_t ws_size,
                              hipStream_t stream) {
  (void)in_sizes; (void)n_in; (void)out_size; (void)ws_size;
  const int H = HIDDEN, V = VOCAB;

  const int*   tok    = (const int*)  d_in[0];
  const float* h0     = (const float*)d_in[1];
  const float* c0     = (const float*)d_in[2];
  const float* enc    = (const float*)d_in[3];
  const float* emb    = (const float*)d_in[4];
  const float* attn_W = (const float*)d_in[5];
  const float* attn_b = (const float*)d_in[6];
  const float* comb_W = (const float*)d_in[7];
  const float* comb_b = (const float*)d_in[8];
  const float* W_ih   = (const float*)d_in[9];
  const float* W_hh   = (const float*)d_in[10];
  const float* b_ih   = (const float*)d_in[11];
  const float* b_hh   = (const float*)d_in[12];
  const float* out_W  = (const float*)d_in[13];
  const float* out_b  = (const float*)d_in[14];

  float* ws    = (float*)d_ws;
  float* xc    = ws;            // 2048: [embedded | attn_applied]
  float* xcmb  = ws + 2048;     // 1024: relu(comb)
  float* gates = ws + 3072;     // 4096
  float* hnew  = ws + 7168;     // 1024
  float* pm    = ws + 8192;     // 64
  float* psum  = ws + 8256;     // 64
  float* lse   = ws + 8320;     // 1

  float* out     = (float*)d_out;
  float* logits  = out;               // logp slot: write logits, fix up in place
  float* h_out   = out + V;
  float* c_out   = out + V + H;
  float* aw_out  = out + V + 2 * H;

  // 1) embedding + attention + context
  attn_kernel<<<1, 256, 0, stream>>>(tok, h0, enc, emb, attn_W, attn_b, xc, aw_out);

  // 2) x = relu(comb_W @ [emb; ctx] + comb_b)      M=1024, K=2048 -> 64 tiles
  wmma_gemv_kernel<<<8, 256, 0, stream>>>(comb_W, xc, comb_b, nullptr,
                                          xcmb, H, 2 * H, 0, 1);

  // 3) gates = W_ih @ x + (b_ih + b_hh)            M=4096, K=1024 -> 256 tiles
  wmma_gemv_kernel<<<32, 256, 0, stream>>>(W_ih, xcmb, b_ih, b_hh,
                                           gates, 4 * H, H, 0, 0);
  // 4) gates += W_hh @ h0
  wmma_gemv_kernel<<<32, 256, 0, stream>>>(W_hh, h0, nullptr, nullptr,
                                           gates, 4 * H, H, 1, 0);

  // 5) LSTM cell elementwise
  lstm_kernel<<<4, 256, 0, stream>>>(gates, c0, h_out, c_out, hnew);

  // 6) logits = out_W @ h_new + out_b              M=50257 -> 3142 tiles
  wmma_gemv_kernel<<<393, 256, 0, stream>>>(out_W, hnew, out_b, nullptr,
                                            logits, V, H, 0, 0);

  // 7) log-softmax: streaming LSE then in-place subtract
  lse_partial_kernel<<<64, 256, 0, stream>>>(logits, V, pm, psum);
  lse_final_kernel<<<1, 64, 0, stream>>>(pm, psum, 64, lse);
  sub_lse_kernel<<<(V + 255) / 256, 256, 0, stream>>>(logits, lse, V);
}